// EnhancedMACDCell_26268019982893
// MI455X (gfx1250) — compile-verified
//
#include <hip/hip_runtime.h>
#include <hip/hip_bf16.h>
#include <math.h>

typedef __attribute__((ext_vector_type(2))) float v2f;
typedef __attribute__((ext_vector_type(8))) float v8f;

#define WAVES 8
#define TPW   4          // tiles (of 16 rows) per wave
#define LROW  68         // padded LDS row stride (floats) -> conflict-free
#define SG_F  68         // floats reserved at LDS front for G + scalars

__device__ __forceinline__ v8f wmma_f32_16x16x4(v2f a, v2f b, v8f c) {
#if defined(__HIP_DEVICE_COMPILE__)
  // 8 args: (neg_a, A, neg_b, B, c_mod, C, reuse_a, reuse_b)
  return __builtin_amdgcn_wmma_f32_16x16x4_f32(false, a, false, b, (short)0, c,
                                               false, false);
#else
  return c;
#endif
}

// ---------------------------------------------------------------------------
// Setup: fold MACD weights into a single 64-tap FIR G[0..63] plus scalars.
//   d_ws layout: G[0..63], [64]=c0, [65]=norm_scale, [66]=norm_bias
// ---------------------------------------------------------------------------
__global__ void macd_setup(const float* __restrict__ wf,  const float* __restrict__ bf,
                           const float* __restrict__ wsl, const float* __restrict__ bsl,
                           const float* __restrict__ wsig,const float* __restrict__ bsig,
                           const float* __restrict__ nsc, const float* __restrict__ nbi,
                           float* __restrict__ g) {
  const int c = threadIdx.x;                 // column 0..63
  float wsum = 0.f;
#pragma unroll
  for (int j = 0; j < 9; ++j) wsum += wsig[j];

  float gc = 0.f;
#pragma unroll
  for (int j = 0; j < 9; ++j) {              // end_j = 56 + j
    const float aj = ((j == 8) ? 1.f : 0.f) - wsig[j];
    float K = 0.f;
    const int idf = c - (44 + j);            // fast window: cols end-12 .. end-1
    if (idf >= 0 && idf < 12) K += wf[idf];
    const int ids = c - (30 + j);            // slow window: cols end-26 .. end-1
    if (ids >= 0 && ids < 26) K -= wsl[ids];
    gc += aj * K;
  }
  g[c] = gc;
  if (c == 0) {
    g[64] = (bf[0] - bsl[0]) * (1.f - wsum) - bsig[0];  // c0
    g[65] = nsc[0];
    g[66] = nbi[0];
  }
}

// ---------------------------------------------------------------------------
// Main: out[b] = tanh((dot(price[b,:], G) + c0) * scale + bias)
// Async global->LDS tile copy (ASYNCcnt), then 16x v_wmma_f32_16x16x4_f32
// per 16-row tile with the G vector broadcast across all 16 B-columns.
// ---------------------------------------------------------------------------
__global__ __launch_bounds__(256) void macd_main(const float* __restrict__ price,
                                                 const float* __restrict__ g,
                                                 float* __restrict__ out,
                                                 int numTiles) {
  // Single LDS block so async-LDS byte offsets are exact (static LDS base = 0):
  // [0..67] = G + scalars, then 8 wave-private 16x68 tiles.
  __shared__ float lds[SG_F + WAVES * 16 * LROW];

  const int tid  = threadIdx.x;
  const int wave = tid >> 5;
  const int lane = tid & 31;

  if (tid < 67) lds[tid] = g[tid];
  __syncthreads();
  const float c0  = lds[64];
  const float nsc = lds[65];
  const float nbi = lds[66];

  const int kb    = (lane >> 4) << 1;   // K sub-offset: 0 (lanes 0-15) or 2 (16-31)
  const int row16 = lane & 15;          // M index within tile
  float* __restrict__ my   = &lds[SG_F + wave * 16 * LROW];
  const unsigned     myOff = (unsigned)(SG_F + wave * 16 * LROW) * 4u; // LDS bytes

  // Hoist loop-invariant B fragments (G broadcast over N) into registers.
  v2f bfrag[16];
#pragma unroll
  for (int i = 0; i < 16; ++i) bfrag[i] = *(const v2f*)&lds[4 * i + kb];

  const long tile0 = (long)blockIdx.x * (WAVES * TPW) + (long)wave * TPW;

  for (int t = 0; t < TPW; ++t) {
    const long tl     = tile0 + t;
    const bool active = tl < (long)numTiles;   // wave-uniform

    if (active) {
      const float* src = price + tl * (16L * 64L);
      if (t + 1 < TPW && tl + 1 < (long)numTiles)
        __builtin_prefetch(src + 1024 + lane * 32, 0, 0);   // next 4KB tile

      // Async copy 16x64 f32 tile directly into padded LDS rows (no VGPR staging).
#pragma unroll
      for (int i = 0; i < 8; ++i) {
        const int f  = i * 32 + lane;           // 0..255 float4s
        const int r  = f >> 4;
        const int c4 = f & 15;
        const unsigned           ldsOff = myOff + (unsigned)(r * LROW + c4 * 4) * 4u;
        const unsigned long long ga     = (unsigned long long)(uintptr_t)(src + 4 * f);
        asm volatile("global_load_async_to_lds_b128 %0, %1, off"
                     :: "v"(ldsOff), "v"(ga) : "memory");
      }
      asm volatile("s_wait_asynccnt 0x0" ::: "memory");
    }
    __syncthreads();

    if (active) {
      v8f acc = {0.f, 0.f, 0.f, 0.f, 0.f, 0.f, 0.f, 0.f};
#pragma unroll
      for (int i = 0; i < 16; ++i) {      // K = 64 in steps of 4
        const int k0 = 4 * i + kb;        // even -> 8B-aligned ds_load_b64
        const v2f a = *(const v2f*)&my[row16 * LROW + k0];  // A[M][k0], A[M][k0+1]
        acc = wmma_f32_16x16x4(a, bfrag[i], acc);
      }
      // D[m][n] == y[m] for every n. Lane 0 holds rows 0..7 in acc[0..7],
      // lane 16 holds rows 8..15.
      if (row16 == 0) {
        const long rowBase = tl * 16 + ((long)(lane >> 4) << 3);
#pragma unroll
        for (int r = 0; r < 8; ++r) {
          const float h = acc[r] + c0;
          out[rowBase + r] = tanhf(h * nsc + nbi);
        }
      }
    }
    __syncthreads();
  }
}

extern "C" void kernel_launch(void* const* d_in, const int* in_sizes, int n_in,
                              void* d_out, int out_size, void* d_ws, size_t ws_size,
                              hipStream_t stream) {
  const float* price = (const float*)d_in[0];
  const float* wf    = (const float*)d_in[1];
  const float* bf    = (const float*)d_in[2];
  const float* wsl   = (const float*)d_in[3];
  const float* bsl   = (const float*)d_in[4];
  const float* wsig  = (const float*)d_in[5];
  const float* bsig  = (const float*)d_in[6];
  const float* nsc   = (const float*)d_in[7];
  const float* nbi   = (const float*)d_in[8];
  float* out = (float*)d_out;
  float* g   = (float*)d_ws;   // 67 floats of scratch

  const int rows     = in_sizes[0] / 64;
  const int numTiles = rows / 16;          // B = 1M -> 65536 tiles

  macd_setup<<<1, 64, 0, stream>>>(wf, bf, wsl, bsl, wsig, bsig, nsc, nbi, g);

  const int tilesPerBlock = WAVES * TPW;   // 32
  const int blocks = (numTiles + tilesPerBlock - 1) / tilesPerBlock;
  macd_main<<<blocks, 256, 0, stream>>>(price, g, out, numTiles);
}